// HeightCompression_63599875719739
// MI455X (gfx1250) — compile-verified
//
#include <hip/hip_runtime.h>

typedef float v2f __attribute__((ext_vector_type(2)));
typedef float v4f __attribute__((ext_vector_type(4)));
typedef float v8f __attribute__((ext_vector_type(8)));

#define CHK 512   // known-point chunk staged in LDS (power of two)

__device__ inline v8f wmma_f32_16x16x4(v2f a, v2f b, v8f c) {
  // D = A(16x4) * B(4x16) + C(16x16), fp32, wave32
  return __builtin_amdgcn_wmma_f32_16x16x4_f32(false, a, false, b, (short)0, c,
                                               false, false);
}

__global__ __launch_bounds__(128) void three_nn_head_kernel(
    const float* __restrict__ unknown,
    const float* __restrict__ kn0, const float* __restrict__ ft0, int M0,
    const float* __restrict__ kn1, const float* __restrict__ ft1, int M1,
    const float* __restrict__ kn2p, const float* __restrict__ ft2p, int M2v,
    const float* __restrict__ Wfc, const float* __restrict__ Wcls,
    const float* __restrict__ Wreg,
    float* __restrict__ out_cls, float* __restrict__ out_reg, int N)
{
  // LDS layout (floats): P(64x96)=6144 | Wcat(64x16)=1024 | regionB=8704
  // regionB phase1: sxy 2*CHK=1024 | szn 2*CHK=1024 | lmd 3072 | lmi 3072 | res 512
  // regionB phase2: hbuf 4*16*64=4096 (overlays sxy/szn/lmd)
  __shared__ __align__(16) float smem[6144 + 1024 + 8704];
  float*    Pbuf = smem;
  float*    Wcat = Pbuf + 64 * 96;
  float*    regb = Wcat + 64 * 16;
  v2f*      sxy  = (v2f*)regb;             // {kx, ky} per known point
  v2f*      szn  = (v2f*)(regb + 2 * CHK); // {kz, |k|^2} per known point
  float*    lmd  = regb + 4 * CHK;
  unsigned* lmi  = (unsigned*)(lmd + 3072);
  float*    resb = (float*)(lmi + 3072);
  float*    hbuf = regb;  // phase-2 overlay

  const int tid   = threadIdx.x;
  const int lane  = tid & 31;
  const int wave  = tid >> 5;
  const int half  = lane >> 4;   // 0: K pair {0,1}/rows 0-7 ; 1: K pair {2,3}/rows 8-15
  const int l16   = lane & 15;
  const int qbase = blockIdx.x * 64 + wave * 16;

  // Build Wcat = [Wcls | Wreg | zeros] : 64 x 16
  for (int i = tid; i < 64 * 16; i += 128) {
    int k = i >> 4, n = i & 15;
    float v = 0.f;
    if (n == 0)      v = Wcls[k];
    else if (n < 4)  v = Wreg[k * 3 + (n - 1)];
    Wcat[i] = v;
  }

  // A matrix (16x4): A[m][0..2] = -2*u_m, A[m][3] = 1 (pairs with B row3 = |k|^2)
  // layout: lanes 0-15 hold K={0,1} of row M=l16, lanes 16-31 hold K={2,3}
  const int arow = qbase + l16;
  float ax = unknown[arow * 3 + 0];
  float ay = unknown[arow * 3 + 1];
  float az = unknown[arow * 3 + 2];
  v2f amat;
  if (half == 0) { amat.x = -2.f * ax; amat.y = -2.f * ay; }
  else           { amat.x = -2.f * az; amat.y = 1.f;       }

  // |u|^2 for the 8 C-layout rows this lane owns (rows r + 8*half).
  // The distance loop tracks SHIFTED distances d' = d2 - |u|^2 (same ordering,
  // saves 8 v_add_f32 per tile); |u|^2 is added back at the dump stage.
  float un2v[8];
  #pragma unroll
  for (int r = 0; r < 8; ++r) {
    int rr = qbase + r + 8 * half;
    float x = unknown[rr * 3 + 0], y = unknown[rr * 3 + 1], z = unknown[rr * 3 + 2];
    un2v[r] = x * x + y * y + z * z;
  }

  // per-half base pointer for the single-b64 B-fragment load
  const v2f* bsrc = (half == 0) ? sxy : szn;

  for (int s = 0; s < 3; ++s) {
    const float* kset = (s == 0) ? kn0 : ((s == 1) ? kn1 : kn2p);
    const float* fset = (s == 0) ? ft0 : ((s == 1) ? ft1 : ft2p);
    const int    M    = (s == 0) ? M0  : ((s == 1) ? M1  : M2v);

    // per-lane running top-3 (per 8 rows), shifted space
    float    bd0[8], bd1[8], bd2[8];
    unsigned bi0[8], bi1[8], bi2[8];
    #pragma unroll
    for (int r = 0; r < 8; ++r) {
      bd0[r] = bd1[r] = bd2[r] = 3.0e38f;
      bi0[r] = bi1[r] = bi2[r] = 0u;
    }

    for (int cb = 0; cb < M; cb += CHK) {
      __syncthreads();
      for (int j = tid; j < CHK; j += 128) {
        int g = cb + j;
        float x, y, z;
        if (g < M) { x = kset[g*3+0]; y = kset[g*3+1]; z = kset[g*3+2]; }
        else       { x = 1.0e17f; y = 0.f; z = 0.f; }   // pad -> huge d2, never wins
        v2f p0; p0.x = x; p0.y = y;
        v2f p1; p1.x = z; p1.y = x*x + y*y + z*z;
        sxy[j] = p0;
        szn[j] = p1;
        int gn = g + CHK;                 // warm next chunk into cache
        if (gn < M) __builtin_prefetch(kset + gn * 3, 0, 0);
      }
      __syncthreads();

      // software-pipelined B-fragment, prefetch distance 2:
      // the fragment consumed by tile t was loaded at tile t-32, so the wait
      // at the WMMA only needs dscnt<=1 and LDS latency is fully hidden.
      v2f b0 = bsrc[l16];
      v2f b1 = bsrc[16 + l16];
      #pragma unroll 2
      for (int t = 0; t < CHK; t += 16) {
        v2f bn = bsrc[((t + 32) & (CHK - 1)) + l16];

        v8f acc = {0.f, 0.f, 0.f, 0.f, 0.f, 0.f, 0.f, 0.f};
        acc = wmma_f32_16x16x4(amat, b0, acc);   // -2 u.k + |k|^2  (= d2 - |u|^2)

        unsigned gi = (unsigned)(cb + t + l16);
        #pragma unroll
        for (int r = 0; r < 8; ++r) {
          float dd = acc[r];               // shifted squared distance
          // scalar skip: only insert when some lane beats its 3rd-best
          if (__builtin_amdgcn_ballot_w32(dd < bd2[r]) != 0u) {
            bool c2 = dd < bd2[r];
            float nd = c2 ? dd : bd2[r];  unsigned ni = c2 ? gi : bi2[r];
            bool c1 = nd < bd1[r];
            float t1 = bd1[r];            unsigned u1 = bi1[r];
            bd1[r] = c1 ? nd : t1;  bi1[r] = c1 ? ni : u1;
            bd2[r] = c1 ? t1 : nd;  bi2[r] = c1 ? u1 : ni;
            bool c0 = bd1[r] < bd0[r];
            float t0 = bd0[r];            unsigned u0 = bi0[r];
            bd0[r] = c0 ? bd1[r] : t0;  bi0[r] = c0 ? bi1[r] : u0;
            bd1[r] = c0 ? t0 : bd1[r];  bi1[r] = c0 ? u0 : bi1[r];
          }
        }
        b0 = b1;
        b1 = bn;
      }
    }

    // dump per-lane top-3 to LDS, un-shifting back to true d2
    __syncthreads();
    #pragma unroll
    for (int r = 0; r < 8; ++r) {
      int row16 = r + 8 * half;
      int base  = (wave * 16 + row16) * 48 + l16 * 3;
      lmd[base + 0] = bd0[r] + un2v[r];
      lmd[base + 1] = bd1[r] + un2v[r];
      lmd[base + 2] = bd2[r] + un2v[r];
      lmi[base + 0] = bi0[r]; lmi[base + 1] = bi1[r]; lmi[base + 2] = bi2[r];
    }
    __syncthreads();

    // merge: one lane per query row scans its 48 candidates -> final top-3 + weights
    if (lane < 16) {
      float m0 = 3e38f, m1 = 3e38f, m2 = 3e38f;
      unsigned j0 = 0, j1 = 0, j2 = 0;
      int base = (wave * 16 + lane) * 48;
      for (int e = 0; e < 48; ++e) {
        float dd = lmd[base + e]; unsigned gi = lmi[base + e];
        bool c2 = dd < m2;
        float nd = c2 ? dd : m2;  unsigned ni = c2 ? gi : j2;
        bool c1 = nd < m1;
        float t1 = m1;            unsigned u1 = j1;
        m1 = c1 ? nd : t1;  j1 = c1 ? ni : u1;
        m2 = c1 ? t1 : nd;  j2 = c1 ? u1 : ni;
        bool c0 = m1 < m0;
        float t0 = m0;            unsigned u0 = j0;
        m0 = c0 ? m1 : t0;  j0 = c0 ? j1 : u0;
        m1 = c0 ? t0 : m1;  j1 = c0 ? u0 : j1;
      }
      float w0 = 1.f / (m0 + 1e-8f);
      float w1 = 1.f / (m1 + 1e-8f);
      float w2 = 1.f / (m2 + 1e-8f);
      float inv = 1.f / (w0 + w1 + w2);
      int rb = (wave * 16 + lane) * 8;
      resb[rb + 0] = w0 * inv; resb[rb + 1] = w1 * inv; resb[rb + 2] = w2 * inv;
      resb[rb + 3] = __uint_as_float(j0);
      resb[rb + 4] = __uint_as_float(j1);
      resb[rb + 5] = __uint_as_float(j2);
    }
    __syncthreads();

    // gather + weighted sum of features -> Pbuf[:, s*32 : s*32+32]
    // feature rows are 128B-aligned 32-float rows: use b128 vector loads/stores
    {
      int row   = lane >> 1;
      int cbase = (lane & 1) * 16;
      int rb    = (wave * 16 + row) * 8;
      float w0 = resb[rb + 0], w1 = resb[rb + 1], w2 = resb[rb + 2];
      unsigned j0 = __float_as_uint(resb[rb + 3]);
      unsigned j1 = __float_as_uint(resb[rb + 4]);
      unsigned j2 = __float_as_uint(resb[rb + 5]);
      const v4f* f0 = (const v4f*)(fset + (size_t)j0 * 32 + cbase);
      const v4f* f1 = (const v4f*)(fset + (size_t)j1 * 32 + cbase);
      const v4f* f2 = (const v4f*)(fset + (size_t)j2 * 32 + cbase);
      v4f* prow = (v4f*)(Pbuf + (wave * 16 + row) * 96 + s * 32 + cbase);
      #pragma unroll
      for (int c = 0; c < 4; ++c)
        prow[c] = w0 * f0[c] + w1 * f1[c] + w2 * f2[c];
    }
    __syncthreads();
  }

  // ---- phase 2a: H(16x64) = P(16x96) @ Wfc(96x64), f32 WMMA K=96 ----
  {
    const int koff = half << 1;
    const int prow = (wave * 16 + l16) * 96;
    #pragma unroll
    for (int nb = 0; nb < 4; ++nb) {
      v8f acc = {0.f, 0.f, 0.f, 0.f, 0.f, 0.f, 0.f, 0.f};
      const int ncol = nb * 16 + l16;
      for (int k = 0; k < 96; k += 4) {
        int kk = k + koff;
        v2f a2 = *(const v2f*)&Pbuf[prow + kk];          // one ds_load_b64
        v2f b2; b2.x = Wfc[kk * 64 + ncol];   b2.y = Wfc[(kk + 1) * 64 + ncol];
        acc = wmma_f32_16x16x4(a2, b2, acc);
      }
      #pragma unroll
      for (int r = 0; r < 8; ++r)
        hbuf[wave * 1024 + (r + 8 * half) * 64 + nb * 16 + l16] = acc[r];
    }
  }
  __syncthreads();

  // ---- phase 2b: O(16x16) = H(16x64) @ Wcat(64x16); cols 0..3 = [cls|reg] ----
  {
    const int koff = half << 1;
    const int hrow = wave * 1024 + l16 * 64;
    v8f acc = {0.f, 0.f, 0.f, 0.f, 0.f, 0.f, 0.f, 0.f};
    for (int k = 0; k < 64; k += 4) {
      int kk = k + koff;
      v2f a2 = *(const v2f*)&hbuf[hrow + kk];            // one ds_load_b64
      v2f b2; b2.x = Wcat[kk * 16 + l16];  b2.y = Wcat[(kk + 1) * 16 + l16];
      acc = wmma_f32_16x16x4(a2, b2, acc);
    }
    #pragma unroll
    for (int r = 0; r < 8; ++r) {
      int orow = qbase + r + 8 * half;
      if (orow < N) {
        if (l16 == 0)      out_cls[orow] = acc[r];
        else if (l16 < 4)  out_reg[orow * 3 + (l16 - 1)] = acc[r];
      }
    }
  }
}

extern "C" void kernel_launch(void* const* d_in, const int* in_sizes, int n_in,
                              void* d_out, int out_size, void* d_ws, size_t ws_size,
                              hipStream_t stream) {
  (void)n_in; (void)d_ws; (void)ws_size; (void)out_size;
  const float* unknown = (const float*)d_in[0];
  const float* kn2 = (const float*)d_in[1];
  const float* ft2 = (const float*)d_in[2];
  const float* kn3 = (const float*)d_in[3];
  const float* ft3 = (const float*)d_in[4];
  const float* kn4 = (const float*)d_in[5];
  const float* ft4 = (const float*)d_in[6];
  const float* Wfc  = (const float*)d_in[7];
  const float* Wcls = (const float*)d_in[8];
  const float* Wreg = (const float*)d_in[9];

  const int N  = in_sizes[0] / 3;
  const int M2 = in_sizes[1] / 3;
  const int M3 = in_sizes[3] / 3;
  const int M4 = in_sizes[5] / 3;

  float* out_cls = (float*)d_out;       // (N,1)
  float* out_reg = out_cls + N;         // (N,3)

  dim3 grid((unsigned)((N + 63) / 64)), block(128);
  three_nn_head_kernel<<<grid, block, 0, stream>>>(
      unknown, kn2, ft2, M2, kn3, ft3, M3, kn4, ft4, M4,
      Wfc, Wcls, Wreg, out_cls, out_reg, N);
}